// GAT_Net_12300786335806
// MI455X (gfx1250) — compile-verified
//
#include <hip/hip_runtime.h>
#include <math.h>

typedef float v2f __attribute__((ext_vector_type(2)));
typedef float v8f __attribute__((ext_vector_type(8)));

#define LRELU_SLOPE 0.2f

// ---------------- order-preserving float<->uint for atomicMax --------------
__device__ __forceinline__ unsigned f2ord(float f) {
  unsigned u = __float_as_uint(f);
  return (u & 0x80000000u) ? ~u : (u | 0x80000000u);
}
__device__ __forceinline__ float ord2f(unsigned u) {
  return __uint_as_float((u & 0x80000000u) ? (u & 0x7FFFFFFFu) : ~u);
}

// ------------- WMMA GEMM: Y[M,Nc] = X[M,K] @ W[K,Nc], all fp32 -------------
// One wave per 16x16 output tile. V_WMMA_F32_16X16X4_F32, K stepped by 4.
// A-frag (16x4): lane = 16*(K/2)+M, vgpr = K%2  (ISA 7.12.2)
// B-frag (4x16): lane = 16*(K/2)+N, vgpr = K%2
// C/D (16x16):   vgpr r -> M = r + 8*(lane/16), N = lane%16
__global__ __launch_bounds__(256) void gemm_wmma_f32(
    const float* __restrict__ X, const float* __restrict__ W,
    float* __restrict__ Y, int M, int K, int Nc) {
  const int lane  = threadIdx.x & 31;
  const int wave  = (blockIdx.x * blockDim.x + threadIdx.x) >> 5;  // wave-uniform
  const int tilesN = Nc >> 4;
  const int tilesM = M >> 4;
  if (wave >= tilesM * tilesN) return;                             // whole-wave exit
  const int mTile = (wave / tilesN) << 4;
  const int nTile = (wave % tilesN) << 4;
  const int half  = lane >> 4;        // 0: K pair base, 1: +2
  const int l16   = lane & 15;

  const float* __restrict__ xrow = X + (size_t)(mTile + l16) * K;
  const float* __restrict__ wcol = W + nTile + l16;

  v8f acc = {0.f, 0.f, 0.f, 0.f, 0.f, 0.f, 0.f, 0.f};
  for (int k = 0; k < K; k += 4) {
    const int ka = k + 2 * half;
    v2f a = *(const v2f*)(xrow + ka);   // 8B-aligned -> global_load_b64
    v2f b;
    b.x = wcol[(size_t)ka * Nc];
    b.y = wcol[(size_t)(ka + 1) * Nc];
#if __has_builtin(__builtin_amdgcn_wmma_f32_16x16x4_f32)
    // 8 args: (neg_a, A, neg_b, B, c_mod, C, reuse_a, reuse_b)
    acc = __builtin_amdgcn_wmma_f32_16x16x4_f32(false, a, false, b,
                                                (short)0, acc, false, false);
#else
    acc[0] += a.x * b.x;  // host pass only
#endif
  }
  float* __restrict__ yrow = Y + nTile + l16;
  const int mbase = mTile + half * 8;
#pragma unroll
  for (int r = 0; r < 8; ++r)
    yrow[(size_t)(mbase + r) * Nc] = acc[r];
}

// --------- attention coefficients: one wave per (node,head), lane = c ------
// C == 32 exactly (both layers), so loads are one coalesced 128B burst and
// the dot products reduce with 5 xor-shuffles (ds_permute on wave32).
__global__ __launch_bounds__(256) void att_coef(
    const float* __restrict__ h, const float* __restrict__ att_s,
    const float* __restrict__ att_d, float* __restrict__ a_s,
    float* __restrict__ a_d, int NH, int H, int C) {
  const long long gt = (long long)blockIdx.x * blockDim.x + threadIdx.x;
  const int wave = (int)(gt >> 5);            // = n*H + head, wave-uniform
  const int c    = threadIdx.x & 31;          // channel
  if (wave >= NH) return;
  const int hh = wave % H;
  const float v = h[(size_t)wave * C + c];
  float ss = v * att_s[hh * C + c];
  float dd = v * att_d[hh * C + c];
#pragma unroll
  for (int off = 16; off > 0; off >>= 1) {
    ss += __shfl_xor(ss, off, 32);
    dd += __shfl_xor(dd, off, 32);
  }
  if (c == 0) { a_s[wave] = ss; a_d[wave] = dd; }
}

// --------------- segment-max of leaky-relu logits over dst -----------------
__global__ __launch_bounds__(256) void edge_max(
    const int* __restrict__ ei, int E, int N, int H,
    const float* __restrict__ a_s, const float* __restrict__ a_d,
    unsigned* __restrict__ m) {
  const int e = blockIdx.x * blockDim.x + threadIdx.x;
  const int Etot = E + N;
  if (e >= Etot) return;
  const int s = (e < E) ? ei[e] : (e - E);       // self loop past E
  const int d = (e < E) ? ei[E + e] : (e - E);
  for (int hh = 0; hh < H; ++hh) {
    float l = a_s[s * H + hh] + a_d[d * H + hh];
    l = (l > 0.f) ? l : LRELU_SLOPE * l;
    atomicMax(&m[d * H + hh], f2ord(l));
  }
}

// ---- accumulate: one wave per (edge,head), lane = channel.  ---------------
// num[d,h,:] += w * h[s,h,:]  (one coalesced 128B atomic burst / wave)
// den[d,h]   += w             (lane 0 only)
__global__ __launch_bounds__(256) void edge_acc(
    const int* __restrict__ ei, int E, int N, int H, int C,
    const float* __restrict__ a_s, const float* __restrict__ a_d,
    const unsigned* __restrict__ m, const float* __restrict__ h,
    float* __restrict__ num, float* __restrict__ den) {
  const long long gt = (long long)blockIdx.x * blockDim.x + threadIdx.x;
  const long long wave = gt >> 5;                // (edge, head), wave-uniform
  const int c = threadIdx.x & 31;                // channel (C == 32)
  const long long Etot = (long long)E + N;
  if (wave >= Etot * H) return;
  const int e  = (int)(wave / H);
  const int hh = (int)(wave % H);
  const int s = (e < E) ? ei[e] : (e - E);       // same addr all lanes -> bcast
  const int d = (e < E) ? ei[E + e] : (e - E);
  float l = a_s[s * H + hh] + a_d[d * H + hh];
  l = (l > 0.f) ? l : LRELU_SLOPE * l;
  const float w = expf(l - ord2f(m[d * H + hh]));
  if (c == 0) atomicAdd(&den[d * H + hh], w);
  const size_t srow = ((size_t)s * H + hh) * C;  // L2-resident gather
  const size_t drow = ((size_t)d * H + hh) * C;
  atomicAdd(&num[drow + c], w * h[srow + c]);
}

// ---------------- out = elu(num/den + bias), elementwise -------------------
__global__ __launch_bounds__(256) void finalize(
    const float* __restrict__ num, const float* __restrict__ den,
    const float* __restrict__ bias, float* __restrict__ out,
    int N, int H, int C) {
  const int t = blockIdx.x * blockDim.x + threadIdx.x;   // n*(H*C)+hh*C+c
  if (t >= N * H * C) return;
  const int nh = t / C;                                  // n*H + hh
  const int hc = t % (H * C);
  float v = num[t] / (den[nh] + 1e-16f) + bias[hc];
  v = (v > 0.f) ? v : (expf(v) - 1.f);                   // ELU (alpha=1)
  out[t] = v;
}

// ---------------------------------------------------------------------------
extern "C" void kernel_launch(void* const* d_in, const int* in_sizes, int n_in,
                              void* d_out, int out_size, void* d_ws, size_t ws_size,
                              hipStream_t stream) {
  const float* x   = (const float*)d_in[0];
  const int*   ei  = (const int*)d_in[1];
  const float* W1  = (const float*)d_in[2];
  const float* as1 = (const float*)d_in[3];
  const float* ad1 = (const float*)d_in[4];
  const float* b1  = (const float*)d_in[5];
  const float* W2  = (const float*)d_in[6];
  const float* as2 = (const float*)d_in[7];
  const float* ad2 = (const float*)d_in[8];
  const float* b2  = (const float*)d_in[9];

  const int HC   = in_sizes[3];                // H*C = 256
  const int Fin  = in_sizes[2] / HC;           // 128
  const int N    = in_sizes[0] / Fin;          // 50000
  const int E    = in_sizes[1] / 2;            // 800000
  const int H    = 8;
  const int C    = HC / H;                     // 32
  const int Cout = in_sizes[7];                // 32
  const int Etot = E + N;

  // ---- workspace carve-up (floats) ----
  float* ws = (float*)d_ws;
  float*    h1   = ws;                               // N*HC
  float*    x2   = h1 + (size_t)N * HC;              // N*HC  (num1, then x2 in place)
  float*    h2   = x2 + (size_t)N * HC;              // N*Cout
  float*    num2 = h2 + (size_t)N * Cout;            // N*Cout
  float*    aS1  = num2 + (size_t)N * Cout;          // N*H
  float*    aD1  = aS1 + (size_t)N * H;              // N*H
  unsigned* m1   = (unsigned*)(aD1 + (size_t)N * H); // N*H
  float*    den1 = (float*)(m1 + (size_t)N * H);     // N*H
  float*    aS2  = den1 + (size_t)N * H;             // N
  float*    aD2  = aS2 + N;                          // N
  unsigned* m2   = (unsigned*)(aD2 + N);             // N
  float*    den2 = (float*)(m2 + N);                 // N

  // init accumulators (0u encodes "below -inf" for the ordered max)
  hipMemsetAsync(x2,   0, (size_t)N * HC   * sizeof(float), stream);
  hipMemsetAsync(num2, 0, (size_t)N * Cout * sizeof(float), stream);
  hipMemsetAsync(m1,   0, (size_t)N * H * sizeof(unsigned), stream);
  hipMemsetAsync(den1, 0, (size_t)N * H * sizeof(float), stream);
  hipMemsetAsync(m2,   0, (size_t)N * sizeof(unsigned), stream);
  hipMemsetAsync(den2, 0, (size_t)N * sizeof(float), stream);

  const int TB = 256;
  auto blocks = [](long long n, int tb) { return (int)((n + tb - 1) / tb); };

  // ================= layer 1 =================
  {
    const long long waves = (long long)(N / 16) * (HC / 16);
    gemm_wmma_f32<<<blocks(waves * 32, TB), TB, 0, stream>>>(x, W1, h1, N, Fin, HC);
    att_coef<<<blocks((long long)N * H * 32, TB), TB, 0, stream>>>(
        h1, as1, ad1, aS1, aD1, N * H, H, C);
    edge_max<<<blocks(Etot, TB), TB, 0, stream>>>(ei, E, N, H, aS1, aD1, m1);
    edge_acc<<<blocks((long long)Etot * H * 32, TB), TB, 0, stream>>>(
        ei, E, N, H, C, aS1, aD1, m1, h1, x2, den1);
    finalize<<<blocks((long long)N * HC, TB), TB, 0, stream>>>(x2, den1, b1, x2, N, H, C);
  }

  // ================= layer 2 (H=1) =================
  {
    const long long waves = (long long)(N / 16) * (Cout / 16);
    gemm_wmma_f32<<<blocks(waves * 32, TB), TB, 0, stream>>>(x2, W2, h2, N, HC, Cout);
    att_coef<<<blocks((long long)N * 32, TB), TB, 0, stream>>>(
        h2, as2, ad2, aS2, aD2, N, 1, Cout);
    edge_max<<<blocks(Etot, TB), TB, 0, stream>>>(ei, E, N, 1, aS2, aD2, m2);
    edge_acc<<<blocks((long long)Etot * 32, TB), TB, 0, stream>>>(
        ei, E, N, 1, Cout, aS2, aD2, m2, h2, num2, den2);
    finalize<<<blocks((long long)N * Cout, TB), TB, 0, stream>>>(
        num2, den2, b2, (float*)d_out, N, 1, Cout);
  }
}